// ChebNet_64991445123445
// MI455X (gfx1250) — compile-verified
//
#include <hip/hip_runtime.h>

typedef float v2f __attribute__((ext_vector_type(2)));
typedef float v8f __attribute__((ext_vector_type(8)));

static constexpr int NN  = 50000;   // nodes
static constexpr int EE  = 800000;  // edges
static constexpr int FIN = 16;      // input features
static constexpr int HH  = 64;      // hidden width
static constexpr int SS  = 7;       // Chebyshev order

// ---------------------------------------------------------------- utilities
__global__ void k_zero_u32(unsigned* __restrict__ p, int n) {
    int i = blockIdx.x * blockDim.x + threadIdx.x;
    if (i < n) p[i] = 0u;
}

__global__ void k_copy_f32(const float* __restrict__ src, float* __restrict__ dst, int n) {
    int i = blockIdx.x * blockDim.x + threadIdx.x;
    if (i < n) dst[i] = src[i];
}

// ---------------------------------------------------------------- CSR build
__global__ void k_count_deg(const int* __restrict__ row, unsigned* __restrict__ deg, int e) {
    int i = blockIdx.x * blockDim.x + threadIdx.x;
    if (i < e) atomicAdd(&deg[row[i]], 1u);
}

__global__ void k_dis(const unsigned* __restrict__ deg, float* __restrict__ dis, int n) {
    int i = blockIdx.x * blockDim.x + threadIdx.x;
    if (i < n) {
        unsigned d = deg[i];
        dis[i] = (d > 0u) ? rsqrtf((float)d) : 0.0f;
    }
}

// single-block exclusive scan (N=50000, 49 chunks of 1024)
__global__ void k_scan(const unsigned* __restrict__ deg, unsigned* __restrict__ rowptr,
                       unsigned* __restrict__ cursor, int n) {
    __shared__ unsigned s[1024];
    int t = threadIdx.x;
    unsigned running = 0;
    for (int base = 0; base < n; base += 1024) {
        unsigned v = (base + t < n) ? deg[base + t] : 0u;
        s[t] = v;
        __syncthreads();
        for (int off = 1; off < 1024; off <<= 1) {
            unsigned mine  = s[t];
            unsigned other = (t >= off) ? s[t - off] : 0u;
            __syncthreads();
            s[t] = mine + other;
            __syncthreads();
        }
        unsigned incl  = s[t];
        unsigned total = s[1023];
        if (base + t < n) {
            unsigned ex = running + incl - v;
            rowptr[base + t] = ex;
            cursor[base + t] = ex;
        }
        running += total;
        __syncthreads();
    }
    if (t == 0) rowptr[n] = running;
}

__global__ void k_fill(const int* __restrict__ row, const int* __restrict__ col,
                       const float* __restrict__ dis, const float* __restrict__ lamp,
                       unsigned* __restrict__ cursor, int* __restrict__ cidx,
                       float* __restrict__ cw, int e) {
    int i = blockIdx.x * blockDim.x + threadIdx.x;
    if (i < e) {
        int r = row[i], c = col[i];
        unsigned p = atomicAdd(&cursor[r], 1u);
        cidx[p] = c;
        cw[p]   = -(2.0f / lamp[0]) * dis[r] * dis[c];
    }
}

// ---------------------------------------------------------------- SpMM: y = alpha*(L_hat x) + beta*xprev
// wave-per-row gather; lanes stride the feature dimension (coalesced)
template <int F>
__global__ void k_spmv(const float* __restrict__ x, const float* __restrict__ xprev,
                       float* __restrict__ y,
                       const unsigned* __restrict__ rowptr, const int* __restrict__ cidx,
                       const float* __restrict__ cw, const float* __restrict__ lamp,
                       float alpha, float beta, int n) {
    int gid  = blockIdx.x * blockDim.x + threadIdx.x;
    int r    = gid >> 5;
    int lane = threadIdx.x & 31;
    if (r >= n) return;

    constexpr int PER = (F + 31) / 32;
    float acc[PER];
#pragma unroll
    for (int p = 0; p < PER; ++p) acc[p] = 0.0f;

    unsigned j0 = rowptr[r], j1 = rowptr[r + 1];
    for (unsigned j = j0; j < j1; ++j) {
        int   c  = cidx[j];
        float wv = cw[j];
        const float* xr = x + (size_t)c * F;
#pragma unroll
        for (int p = 0; p < PER; ++p) {
            int f = lane + 32 * p;
            if (F >= 32 || f < F) acc[p] += wv * xr[f];
        }
    }
    float diag = 2.0f / lamp[0] - 1.0f;
    const float* xi = x + (size_t)r * F;
    const float* xp = xprev + (size_t)r * F;
    float*       yi = y + (size_t)r * F;
#pragma unroll
    for (int p = 0; p < PER; ++p) {
        int f = lane + 32 * p;
        if (F >= 32 || f < F)
            yi[f] = alpha * (acc[p] + diag * xi[f]) + beta * xp[f];
    }
}

// ---------------------------------------------------------------- dense GEMM: Out[N,64] += X[N,K] @ W[K,64]
// block = 128 thr (4 waves). Block owns 16 rows; wave w owns cols [16w,16w+16).
// A tile staged via LDS (stride K+1 kills 64-bank conflicts), fp32 WMMA 16x16x4.
template <int K>
__global__ void k_gemm_wmma(const float* __restrict__ X, const float* __restrict__ W,
                            float* __restrict__ Out) {
    constexpr int LDA = K + 1;
    __shared__ float As[16 * LDA];

    int tid = threadIdx.x;
    int bm  = blockIdx.x;
    const float* Xt = X + (size_t)bm * 16 * K;
    for (int li = tid; li < 16 * K; li += 128) {
        int r = li / K, k = li % K;
        As[r * LDA + k] = Xt[li];
    }
    __syncthreads();

    int wave = tid >> 5;
    int lane = tid & 31;
    int half = lane >> 4;   // 0: lanes 0-15, 1: lanes 16-31
    int lr   = lane & 15;
    int n0   = wave * 16;

    v8f c = {0.f, 0.f, 0.f, 0.f, 0.f, 0.f, 0.f, 0.f};
#pragma unroll
    for (int kk = 0; kk < K; kk += 4) {
        // A 16x4 f32: lanes0-15 hold M=lr, K=kk..kk+1 ; lanes16-31 hold K=kk+2..kk+3
        v2f a, b;
        a.x = As[lr * LDA + kk + 2 * half];
        a.y = As[lr * LDA + kk + 2 * half + 1];
        // B 4x16 f32: lanes = N (cols), VGPRs over K, mirrored halves
        const float* Wp = W + (size_t)(kk + 2 * half) * HH + n0 + lr;
        b.x = Wp[0];
        b.y = Wp[HH];
        c = __builtin_amdgcn_wmma_f32_16x16x4_f32(false, a, false, b, (short)0, c, false, false);
    }
    // C: VGPR r -> M = r (lanes 0-15) / r+8 (lanes 16-31), N = lr
    float* Ot = Out + ((size_t)bm * 16 + half * 8) * HH + n0 + lr;
#pragma unroll
    for (int r = 0; r < 8; ++r) Ot[(size_t)r * HH] += c[r];
}

// ---------------------------------------------------------------- epilogues
__global__ void k_bias_relu(const float* __restrict__ acc, const float* __restrict__ b,
                            float* __restrict__ h, int total) {
    int i = blockIdx.x * blockDim.x + threadIdx.x;
    if (i < total) h[i] = fmaxf(acc[i] + b[i & (HH - 1)], 0.0f);
}

__global__ void k_final(const float* __restrict__ h, const float* __restrict__ Wfc,
                        const float* __restrict__ bfc, float* __restrict__ out, int n) {
    int gid  = blockIdx.x * blockDim.x + threadIdx.x;
    int node = gid >> 5;
    int lane = threadIdx.x & 31;
    if (node >= n) return;
    const float* hr = h + (size_t)node * HH;
    float a = hr[lane] * Wfc[lane] + hr[lane + 32] * Wfc[lane + 32];
    for (int o = 16; o > 0; o >>= 1) a += __shfl_xor(a, o, 32);
    if (lane == 0) out[node] = a + bfc[0];
}

// ---------------------------------------------------------------- host side
extern "C" void kernel_launch(void* const* d_in, const int* in_sizes, int n_in,
                              void* d_out, int out_size, void* d_ws, size_t ws_size,
                              hipStream_t stream) {
    (void)in_sizes; (void)n_in; (void)out_size; (void)ws_size;

    const float* x   = (const float*)d_in[0];
    const int*   row = (const int*)d_in[1];
    const int*   col = row + EE;
    const float* lam = (const float*)d_in[2];
    const float* W1  = (const float*)d_in[3];
    const float* b1  = (const float*)d_in[4];
    const float* W2  = (const float*)d_in[5];
    const float* b2  = (const float*)d_in[6];
    const float* W3  = (const float*)d_in[7];
    const float* b3  = (const float*)d_in[8];
    const float* Wfc = (const float*)d_in[9];
    const float* bfc = (const float*)d_in[10];
    float*       out = (float*)d_out;

    char*  ws  = (char*)d_ws;
    size_t off = 0;
    auto alloc = [&](size_t bytes) {
        char* p = ws + off;
        off += (bytes + 255) & ~(size_t)255;
        return p;
    };
    const size_t NB = (size_t)NN * HH * sizeof(float);
    float*    B0     = (float*)alloc(NB);
    float*    B1     = (float*)alloc(NB);
    float*    B2     = (float*)alloc(NB);
    float*    B3     = (float*)alloc(NB);
    float*    ACC    = (float*)alloc(NB);
    unsigned* deg    = (unsigned*)alloc((size_t)NN * 4);
    float*    dis    = (float*)alloc((size_t)NN * 4);
    unsigned* rowptr = (unsigned*)alloc((size_t)(NN + 1) * 4);
    unsigned* cursor = (unsigned*)alloc((size_t)NN * 4);
    int*      cidx   = (int*)alloc((size_t)EE * 4);
    float*    cw     = (float*)alloc((size_t)EE * 4);

    const int TB = 256;
    const int gN  = (NN + TB - 1) / TB;
    const int gE  = (EE + TB - 1) / TB;
    const int gNH = (NN * HH + TB - 1) / TB;
    const int gW  = (NN * 32 + TB - 1) / TB;   // wave-per-row launches
    const int gT  = NN / 16;                   // 3125 GEMM row-tiles

    // ---- build normalized CSR (once; reused by all 18 SpMMs)
    k_zero_u32<<<gN, TB, 0, stream>>>(deg, NN);
    k_count_deg<<<gE, TB, 0, stream>>>(row, deg, EE);
    k_dis<<<gN, TB, 0, stream>>>(deg, dis, NN);
    k_scan<<<1, 1024, 0, stream>>>(deg, rowptr, cursor, NN);
    k_fill<<<gE, TB, 0, stream>>>(row, col, dis, lam, cursor, cidx, cw, EE);

    // ---- layer 1 (F=16)
    k_copy_f32<<<(NN * FIN + TB - 1) / TB, TB, 0, stream>>>(x, B0, NN * FIN);
    k_zero_u32<<<gNH, TB, 0, stream>>>((unsigned*)ACC, NN * HH);
    k_gemm_wmma<FIN><<<gT, 128, 0, stream>>>(B0, W1, ACC);
    k_spmv<FIN><<<gW, TB, 0, stream>>>(B0, B0, B1, rowptr, cidx, cw, lam, 1.f, 0.f, NN);
    k_gemm_wmma<FIN><<<gT, 128, 0, stream>>>(B1, W1 + 1 * FIN * HH, ACC);
    {
        float *t0 = B0, *t1 = B1, *t2 = B2;
        for (int k = 2; k < SS; ++k) {
            k_spmv<FIN><<<gW, TB, 0, stream>>>(t1, t0, t2, rowptr, cidx, cw, lam, 2.f, -1.f, NN);
            k_gemm_wmma<FIN><<<gT, 128, 0, stream>>>(t2, W1 + k * FIN * HH, ACC);
            float* tmp = t0; t0 = t1; t1 = t2; t2 = tmp;
        }
    }
    k_bias_relu<<<gNH, TB, 0, stream>>>(ACC, b1, B3, NN * HH);   // h1 -> B3

    // ---- layer 2 (F=64)
    k_zero_u32<<<gNH, TB, 0, stream>>>((unsigned*)ACC, NN * HH);
    k_gemm_wmma<HH><<<gT, 128, 0, stream>>>(B3, W2, ACC);
    k_spmv<HH><<<gW, TB, 0, stream>>>(B3, B3, B0, rowptr, cidx, cw, lam, 1.f, 0.f, NN);
    k_gemm_wmma<HH><<<gT, 128, 0, stream>>>(B0, W2 + 1 * HH * HH, ACC);
    {
        float *t0 = B3, *t1 = B0, *t2 = B1;
        for (int k = 2; k < SS; ++k) {
            k_spmv<HH><<<gW, TB, 0, stream>>>(t1, t0, t2, rowptr, cidx, cw, lam, 2.f, -1.f, NN);
            k_gemm_wmma<HH><<<gT, 128, 0, stream>>>(t2, W2 + k * HH * HH, ACC);
            float* tmp = t0; t0 = t1; t1 = t2; t2 = tmp;
        }
    }
    k_bias_relu<<<gNH, TB, 0, stream>>>(ACC, b2, B2, NN * HH);   // h2 -> B2

    // ---- layer 3 (F=64)
    k_zero_u32<<<gNH, TB, 0, stream>>>((unsigned*)ACC, NN * HH);
    k_gemm_wmma<HH><<<gT, 128, 0, stream>>>(B2, W3, ACC);
    k_spmv<HH><<<gW, TB, 0, stream>>>(B2, B2, B0, rowptr, cidx, cw, lam, 1.f, 0.f, NN);
    k_gemm_wmma<HH><<<gT, 128, 0, stream>>>(B0, W3 + 1 * HH * HH, ACC);
    {
        float *t0 = B2, *t1 = B0, *t2 = B1;
        for (int k = 2; k < SS; ++k) {
            k_spmv<HH><<<gW, TB, 0, stream>>>(t1, t0, t2, rowptr, cidx, cw, lam, 2.f, -1.f, NN);
            k_gemm_wmma<HH><<<gT, 128, 0, stream>>>(t2, W3 + k * HH * HH, ACC);
            float* tmp = t0; t0 = t1; t1 = t2; t2 = tmp;
        }
    }
    k_bias_relu<<<gNH, TB, 0, stream>>>(ACC, b3, B3, NN * HH);   // h3 -> B3

    // ---- final projection
    k_final<<<gW, TB, 0, stream>>>(B3, Wfc, bfc, out, NN);
}